// VolumeAwareLoss_25606595018779
// MI455X (gfx1250) — compile-verified
//
#include <hip/hip_runtime.h>
#include <hip/hip_bf16.h>
#include <math.h>

#define SVOX 262144      // 64^3
#define NB 8
#define NC 5
#define NPAIR 32
#define EPSF 1e-5f
#define BIGF 1e8f

typedef __attribute__((ext_vector_type(2))) float v2f;
typedef __attribute__((ext_vector_type(8))) float v8f;

// ---------------- Kernel A: softmax, CE, dice/tversky partials, masks ----------------
// partials layout: [b][block (256)][16]  features: 0-3 inter, 4-7 sum_p, 8 ce, 9-15 zero
// icnt layout: [b][9]  0-4 vol counts (target==c), 5-8 pcount (prob_c>0.5), ints
__global__ __launch_bounds__(256) void kA(const float* __restrict__ pred,
                                          const int* __restrict__ target,
                                          unsigned char* __restrict__ pb,
                                          unsigned char* __restrict__ tb,
                                          float* __restrict__ partials,
                                          int* __restrict__ icnt) {
    const int b   = blockIdx.y;
    const int tid = threadIdx.x;
    const size_t pbase = (size_t)b * NC * SVOX;

    float acc[9];
#pragma unroll
    for (int i = 0; i < 9; ++i) acc[i] = 0.f;
    int vc[5] = {0,0,0,0,0};
    int pc[4] = {0,0,0,0};

    const int v0 = blockIdx.x * 1024;
#pragma unroll
    for (int k = 0; k < 4; ++k) {
        const int v = v0 + k * 256 + tid;
        const int t = target[(size_t)b * SVOX + v];
        float p[5];
#pragma unroll
        for (int c = 0; c < 5; ++c) p[c] = pred[pbase + (size_t)c * SVOX + v];
        float m = p[0];
#pragma unroll
        for (int c = 1; c < 5; ++c) m = fmaxf(m, p[c]);
        float e[5], Z = 0.f;
#pragma unroll
        for (int c = 0; c < 5; ++c) { e[c] = expf(p[c] - m); Z += e[c]; }
        const float inv  = 1.f / Z;
        const float logZ = logf(Z);
        const float pt = (t==0)?p[0]:(t==1)?p[1]:(t==2)?p[2]:(t==3)?p[3]:p[4];
        acc[8] += (m + logZ - pt);            // -log_softmax[target]
#pragma unroll
        for (int cm = 0; cm < 4; ++cm) {
            const float prob = e[cm + 1] * inv;
            const bool  g    = (t == cm + 1);
            acc[cm]     += g ? prob : 0.f;
            acc[4 + cm] += prob;
            const bool pbit = prob > 0.5f;
            pb[(size_t)(b * 4 + cm) * SVOX + v] = pbit ? 1 : 0;
            tb[(size_t)(b * 4 + cm) * SVOX + v] = g    ? 1 : 0;
            if (pbit) pc[cm]++;
        }
        vc[t]++;
    }

    __shared__ float red[256];
    float* out = partials + ((size_t)b * 256 + blockIdx.x) * 16;
    for (int f = 0; f < 9; ++f) {
        red[tid] = acc[f];
        __syncthreads();
        for (int s = 128; s > 0; s >>= 1) {
            if (tid < s) red[tid] += red[tid + s];
            __syncthreads();
        }
        if (tid == 0) out[f] = red[0];
        __syncthreads();
    }
    if (tid == 0) { for (int f = 9; f < 16; ++f) out[f] = 0.f; }

    __shared__ int ired[9];
    if (tid < 9) ired[tid] = 0;
    __syncthreads();
#pragma unroll
    for (int c = 0; c < 5; ++c) atomicAdd(&ired[c], vc[c]);
#pragma unroll
    for (int cm = 0; cm < 4; ++cm) atomicAdd(&ired[5 + cm], pc[cm]);
    __syncthreads();
    if (tid < 9) atomicAdd(&icnt[b * 9 + tid], ired[tid]);
}

// ---------------- Stage-2 reduction with f32 WMMA (one wave per batch) ----------------
// D = A x ones accumulation: A[m,k] = partials[4t+k][feature m]; after 64 iters
// D[m,*] = sum over 256 blocks of feature m. Full f32 precision, fixed order.
__global__ __launch_bounds__(32) void kRedWMMA(const float* __restrict__ partials,
                                               float* __restrict__ bsums) {
    const int b    = blockIdx.x;
    const int lane = threadIdx.x;
    const float* P = partials + (size_t)b * 256 * 16;
    v8f c = {0.f,0.f,0.f,0.f,0.f,0.f,0.f,0.f};
    v2f ones = {1.f, 1.f};
    const int f    = lane & 15;
    const int half = (lane >> 4) << 1;     // lanes 0-15 -> K 0/1, lanes 16-31 -> K 2/3
    for (int t = 0; t < 64; ++t) {
        const int blk = 4 * t + half;
        v2f a;
        a[0] = P[(size_t)blk * 16 + f];
        a[1] = P[(size_t)(blk + 1) * 16 + f];
        c = __builtin_amdgcn_wmma_f32_16x16x4_f32(false, a, false, ones,
                                                  (short)0, c, false, false);
    }
    // D[m,n]: lanes 0-15 VGPR r = D[r, lane]; lanes 16-31 VGPR r = D[r+8, lane-16]
    if (lane == 0) {
#pragma unroll
        for (int m = 0; m < 8; ++m) bsums[b * 16 + m] = c[m];
    } else if (lane == 16) {
#pragma unroll
        for (int m = 0; m < 8; ++m) bsums[b * 16 + 8 + m] = c[m];
    }
}

// ---------------- Finalize 1: class weights, dice, tversky, CE ----------------
__global__ void kFin1(const float* __restrict__ bsums, const int* __restrict__ icnt,
                      float* __restrict__ wdwt, float* __restrict__ dout) {
    if (threadIdx.x != 0) return;
    const float SW[5] = {1.f, 2.f, 1.5f, 2.5f, 1.5f};
    const float BV[5] = {0.f, 2000.f, 8000.f, 1000.f, 3000.f};
    float sumw[NB];
    for (int b = 0; b < NB; ++b) {
        float eff[5], s = 0.f;
        for (int c = 0; c < 5; ++c) {
            const float vol  = (float)icnt[b * 9 + c];
            float mult = sqrtf(BV[c] / fmaxf(vol, EPSF));
            mult = fminf(mult, 3.f);
            eff[c] = SW[c] * mult;
            s += eff[c];
        }
        const float norm = s / 5.f;
        float sw = 0.f;
        for (int c = 0; c < 5; ++c) {
            const float wv = eff[c] / fmaxf(norm, EPSF);
            dout[12 + b * 5 + c] = wv;     // w output
            sw += wv;
        }
        sumw[b] = sw;
    }
    float dsum = 0.f, tsum = 0.f, cesum = 0.f;
    for (int b = 0; b < NB; ++b) {
        cesum += bsums[b * 16 + 8];
        for (int cm = 0; cm < 4; ++cm) {
            const float inter = bsums[b * 16 + cm];
            const float sump  = bsums[b * 16 + 4 + cm];
            const float sumg  = (float)icnt[b * 9 + cm + 1];
            dsum += 1.f - (2.f * inter + EPSF) / ((sump + sumg) + EPSF);
            const float fp = sump - inter, fn = sumg - inter;
            tsum += 1.f - (inter + EPSF) / (inter + 0.3f * fp + 0.7f * fn + EPSF);
        }
    }
    const float dice    = dsum / 32.f;
    const float ce      = cesum / (8.f * (float)SVOX);
    const float dice_ce = 0.5f * dice + 0.5f * ce;
    const float tv      = tsum / 32.f;
    for (int b = 0; b < NB; ++b) {
        wdwt[b]     = dice_ce * sumw[b];   // weighted_dice[b]
        wdwt[8 + b] = tv * sumw[b];        // weighted_tversky[b]
    }
}

// ---------------- Surface extraction (m & ~erode6(m)) + surface counts ----------------
__global__ __launch_bounds__(256) void kSurf(const unsigned char* __restrict__ msk,
                                             unsigned char* __restrict__ surf,
                                             int* __restrict__ cnt) {
    const int pair = blockIdx.y;
    const size_t base = (size_t)pair * SVOX;
    const int tid = threadIdx.x;
    int local = 0;
#pragma unroll
    for (int k = 0; k < 4; ++k) {
        const int v = blockIdx.x * 1024 + k * 256 + tid;
        const unsigned char m = msk[base + v];
        unsigned char s = 0;
        if (m) {
            const int x = v & 63, y = (v >> 6) & 63, z = v >> 12;
            bool er = (x > 0)  && msk[base + v - 1];
            er = er && (x < 63) && msk[base + v + 1];
            er = er && (y > 0)  && msk[base + v - 64];
            er = er && (y < 63) && msk[base + v + 64];
            er = er && (z > 0)  && msk[base + v - 4096];
            er = er && (z < 63) && msk[base + v + 4096];
            s = er ? 0 : 1;
        }
        surf[base + v] = s;
        local += s;
    }
    __shared__ int sred;
    if (tid == 0) sred = 0;
    __syncthreads();
    atomicAdd(&sred, local);
    __syncthreads();
    if (tid == 0) atomicAdd(&cnt[pair], sred);
}

// ---------------- EDT pass X (seeds mask ? 0 : BIG, min-plus along x) ----------------
__global__ __launch_bounds__(256) void kEdtX(const unsigned char* __restrict__ surf,
                                             float* __restrict__ f) {
    const int pair = blockIdx.y;
    const int r = threadIdx.x >> 6;          // local row 0..3
    const int i = threadIdx.x & 63;
    const int row = blockIdx.x * 4 + r;      // 0..4095
    __shared__ float line[4][64];
    const size_t base = (size_t)pair * SVOX + (size_t)row * 64;
    line[r][i] = surf[base + i] ? 0.f : BIGF;
    __syncthreads();
    float best = 3.4e38f;
    for (int j = 0; j < 64; ++j) {
        const int d = i - j;
        best = fminf(best, line[r][j] + (float)(d * d));
    }
    f[base + i] = best;
}

// ---------------- EDT pass Y (per z-plane tile in LDS) ----------------
__global__ __launch_bounds__(256) void kEdtY(float* __restrict__ f) {
    const int pair = blockIdx.y, z = blockIdx.x;
    __shared__ float t[64][64];              // [y][x]
    const size_t base = (size_t)pair * SVOX + (size_t)z * 4096;
#pragma unroll
    for (int k = 0; k < 16; ++k) {
        const int l = threadIdx.x + 256 * k;
        t[l >> 6][l & 63] = f[base + l];
    }
    __syncthreads();
    const int x = threadIdx.x & 63, y0 = threadIdx.x >> 6;
    float res[16];
#pragma unroll
    for (int k = 0; k < 16; ++k) {
        const int y = y0 + 4 * k;
        float best = 3.4e38f;
        for (int j = 0; j < 64; ++j) {
            const int d = y - j;
            best = fminf(best, t[j][x] + (float)(d * d));
        }
        res[k] = best;
    }
#pragma unroll
    for (int k = 0; k < 16; ++k)
        f[base + (size_t)(y0 + 4 * k) * 64 + x] = res[k];
}

// ---------------- EDT pass Z (per y-slab tile in LDS) ----------------
__global__ __launch_bounds__(256) void kEdtZ(float* __restrict__ f) {
    const int pair = blockIdx.y, y = blockIdx.x;
    __shared__ float t[64][64];              // [z][x]
    const size_t base = (size_t)pair * SVOX + (size_t)y * 64;
#pragma unroll
    for (int k = 0; k < 16; ++k) {
        const int l = threadIdx.x + 256 * k;
        t[l >> 6][l & 63] = f[base + (size_t)(l >> 6) * 4096 + (l & 63)];
    }
    __syncthreads();
    const int x = threadIdx.x & 63, z0 = threadIdx.x >> 6;
    float res[16];
#pragma unroll
    for (int k = 0; k < 16; ++k) {
        const int z = z0 + 4 * k;
        float best = 3.4e38f;
        for (int j = 0; j < 64; ++j) {
            const int d = z - j;
            best = fminf(best, t[j][x] + (float)(d * d));
        }
        res[k] = best;
    }
#pragma unroll
    for (int k = 0; k < 16; ++k)
        f[base + (size_t)(z0 + 4 * k) * 4096 + x] = res[k];
}

// ---------------- Masked sqrt-sum: sum over src_surf of sqrt(edt) ----------------
__global__ __launch_bounds__(256) void kSum(const float* __restrict__ f,
                                            const unsigned char* __restrict__ srcs,
                                            float* __restrict__ part) {
    const int pair = blockIdx.y;
    const size_t base = (size_t)pair * SVOX;
    const int tid = threadIdx.x;
    float local = 0.f;
#pragma unroll
    for (int k = 0; k < 16; ++k) {
        const int v = blockIdx.x * 4096 + 256 * k + tid;
        if (srcs[base + v]) local += sqrtf(f[base + v]);
    }
    __shared__ float red[256];
    red[tid] = local;
    __syncthreads();
    for (int s = 128; s > 0; s >>= 1) {
        if (tid < s) red[tid] += red[tid + s];
        __syncthreads();
    }
    if (tid == 0) part[pair * 64 + blockIdx.x] = red[0];
}

__global__ __launch_bounds__(64) void kSum2(const float* __restrict__ part,
                                            float* __restrict__ out) {
    const int pair = blockIdx.x;
    const int tid = threadIdx.x;
    __shared__ float red[64];
    red[tid] = part[pair * 64 + tid];
    __syncthreads();
    for (int s = 32; s > 0; s >>= 1) {
        if (tid < s) red[tid] += red[tid + s];
        __syncthreads();
    }
    if (tid == 0) out[pair] = red[0];
}

// ---------------- Finalize 2: surface loss, totals, means ----------------
__global__ void kFin2(const float* __restrict__ SA, const float* __restrict__ SB,
                      const int* __restrict__ cntps, const int* __restrict__ cntts,
                      const int* __restrict__ icnt, const float* __restrict__ wdwt,
                      float* __restrict__ dout) {
    if (threadIdx.x != 0) return;
    float sl[NB];
    for (int b = 0; b < NB; ++b) sl[b] = 0.f;
    for (int pair = 0; pair < NPAIR; ++pair) {
        const int b = pair >> 2, cm = pair & 3;
        const float msdA = SA[pair] / fmaxf((float)cntts[pair], 1.f); // msd(ts -> ps)
        const float msdB = SB[pair] / fmaxf((float)cntps[pair], 1.f); // msd(ps -> ts)
        const float ms = 0.5f * (msdA + msdB);
        const bool tany = icnt[b * 9 + cm + 1] > 0;
        const bool pany = icnt[b * 9 + 5 + cm] > 0;
        const float wf = dout[12 + b * 5 + (cm + 1)];
        const float contrib = (!tany && !pany) ? 0.f
                             : ((tany != pany) ? wf * 10.f : wf * ms);
        sl[b] += contrib;
    }
    float tsum = 0.f, wdsum = 0.f, wtsum = 0.f, slsum = 0.f;
    for (int b = 0; b < NB; ++b) {
        sl[b] *= 0.25f;
        const float tot = 0.4f * wdwt[b] + 0.4f * wdwt[8 + b] + 0.2f * sl[b];
        dout[4 + b] = tot;
        tsum += tot; wdsum += wdwt[b]; wtsum += wdwt[8 + b]; slsum += sl[b];
    }
    dout[0] = tsum * 0.125f;
    dout[1] = wdsum * 0.125f;
    dout[2] = wtsum * 0.125f;
    dout[3] = slsum * 0.125f;
}

// ---------------- Host launch ----------------
extern "C" void kernel_launch(void* const* d_in, const int* in_sizes, int n_in,
                              void* d_out, int out_size, void* d_ws, size_t ws_size,
                              hipStream_t stream) {
    const float* pred   = (const float*)d_in[0];
    const int*   target = (const int*)d_in[1];
    float* dout = (float*)d_out;
    char* ws = (char*)d_ws;

    // workspace layout (EDT buffer overlaps pb/tb which die after kSurf)
    float*         edt    = (float*)(ws + 0);                    // 32 MB
    unsigned char* pb     = (unsigned char*)(ws + 0);            //  8 MB
    unsigned char* tb     = (unsigned char*)(ws + 8388608);      //  8 MB
    unsigned char* ps     = (unsigned char*)(ws + 33554432);     //  8 MB
    unsigned char* ts     = (unsigned char*)(ws + 41943040);     //  8 MB
    float*         partA  = (float*)(ws + 50331648);             // 128 KB
    float*         bsums  = (float*)(ws + 50462720);             // 512 B
    float*         partS  = (float*)(ws + 50463232);             // 8 KB
    float*         SA     = (float*)(ws + 50471424);             // 128 B
    float*         SB     = (float*)(ws + 50471552);             // 128 B
    float*         wdwt   = (float*)(ws + 50471680);             // 64 B
    int*           icnt   = (int*)(ws + 50471744);               // 72 ints
    int*           cntps  = icnt + 72;                           // 32 ints
    int*           cntts  = icnt + 104;                          // 32 ints

    hipMemsetAsync(icnt, 0, 136 * sizeof(int), stream);

    kA<<<dim3(256, 8), 256, 0, stream>>>(pred, target, pb, tb, partA, icnt);
    kRedWMMA<<<8, 32, 0, stream>>>(partA, bsums);
    kFin1<<<1, 1, 0, stream>>>(bsums, icnt, wdwt, dout);

    kSurf<<<dim3(256, 32), 256, 0, stream>>>(pb, ps, cntps);
    kSurf<<<dim3(256, 32), 256, 0, stream>>>(tb, ts, cntts);

    // EDT(ps), then sum sqrt over ts
    kEdtX<<<dim3(1024, 32), 256, 0, stream>>>(ps, edt);
    kEdtY<<<dim3(64, 32), 256, 0, stream>>>(edt);
    kEdtZ<<<dim3(64, 32), 256, 0, stream>>>(edt);
    kSum<<<dim3(64, 32), 256, 0, stream>>>(edt, ts, partS);
    kSum2<<<32, 64, 0, stream>>>(partS, SA);

    // EDT(ts), then sum sqrt over ps
    kEdtX<<<dim3(1024, 32), 256, 0, stream>>>(ts, edt);
    kEdtY<<<dim3(64, 32), 256, 0, stream>>>(edt);
    kEdtZ<<<dim3(64, 32), 256, 0, stream>>>(edt);
    kSum<<<dim3(64, 32), 256, 0, stream>>>(edt, ps, partS);
    kSum2<<<32, 64, 0, stream>>>(partS, SB);

    kFin2<<<1, 1, 0, stream>>>(SA, SB, cntps, cntts, icnt, wdwt, dout);
}